// Dist_Conv2D_11802570129555
// MI455X (gfx1250) — compile-verified
//
#include <hip/hip_runtime.h>

#ifndef __has_builtin
#define __has_builtin(x) 0
#endif

#if __has_builtin(__builtin_amdgcn_global_load_async_to_lds_b32)
#define USE_ASYNC_LDS 1
#else
#define USE_ASYNC_LDS 0
#endif

// Problem constants
#define CIN      64
#define HW       56
#define COUT     128
#define DTOT     576          // 64*3*3, d = c*9 + kh*3 + kw (channel-major, matches reference)
#define BATCH    16

// Tiling
#define OCB      32           // output channels per workgroup
#define TROWS    4            // output rows per workgroup
#define TCOLS    28           // output cols per workgroup (56 = 2*28)
#define NTHREADS 224          // 8 channel-groups x (4 rows x 7 col-groups), 7 waves of 32

// LDS layout (floats)
#define XR_STR   32                 // padded row stride (30 used)
#define XC_STR   (6 * XR_STR)       // 192: 6 halo rows per channel
#define XLDS_N   (CIN * XC_STR)     // 12288 floats = 48 KB
#define WLDS_N   (DTOT * OCB)       // 18432 floats = 72 KB, layout [d][oc']
#define LDS_BYTES ((XLDS_N + WLDS_N) * 4)

#if USE_ASYNC_LDS
// Builtin prototype (per hipcc diagnostic): AS1 int* global src, AS3 int* LDS dst,
// imm offset, imm cpol.
typedef __attribute__((address_space(1))) int as1_int;
typedef __attribute__((address_space(3))) int as3_int;

__device__ __forceinline__ void async_g2l_b32(const float* g, float* l) {
  // per-lane async DMA: copies 4 bytes from the global address in g to the LDS address in l; tracked by ASYNCcnt, EXEC-masked
  __builtin_amdgcn_global_load_async_to_lds_b32((as1_int*)g, (as3_int*)l, 0, 0);
}
__device__ __forceinline__ void wait_async0() {
#if __has_builtin(__builtin_amdgcn_s_wait_asynccnt)
  __builtin_amdgcn_s_wait_asynccnt(0);
#else
  asm volatile("s_wait_asynccnt 0x0" ::: "memory");
#endif
}
#endif

__global__ __launch_bounds__(NTHREADS) void cheb_conv_kernel(
    const float* __restrict__ Xg,    // (16,64,56,56)
    const float* __restrict__ Wg,    // (128,64,3,3) == (128,576)
    const float* __restrict__ Bias,  // (128)
    float* __restrict__ Out)         // (16,128,56,56)
{
  extern __shared__ float smem[];
  float* Xlds = smem;            // [64][6][XR_STR]
  float* Wlds = smem + XLDS_N;   // [576][32]  (transposed weight tile)

  const int tid  = threadIdx.x;
  const int og   = tid / 28;          // 0..7  : channel group (4 channels)
  const int pg   = tid % 28;
  const int row  = pg / 7;            // 0..3  : output row within tile
  const int colb = (pg % 7) * 4;      // 0..24 : first of 4 output cols

  const int ocb = blockIdx.x & 3;     // 0..3 channel blocks
  const int ct  = blockIdx.x >> 2;    // 0..1 col tiles
  const int rt  = blockIdx.y;         // 0..13 row tiles
  const int b   = blockIdx.z;         // 0..15 batch

  const int rowbase = rt * TROWS;
  const int colbase = ct * TCOLS;
  const int ocbase  = ocb * OCB;

  // ---------------- stage: zero x halo tile ----------------
  for (int i = tid; i < XLDS_N; i += NTHREADS) Xlds[i] = 0.0f;
  __syncthreads();  // DS stores visible before async engine writes same addrs

  // ---------------- stage: x tile (64ch x 6rows x 30cols, halo=1) ----------------
  const float* Xb = Xg + (size_t)b * CIN * HW * HW;
  for (int i = tid; i < CIN * 6 * 30; i += NTHREADS) {
    const int c  = i / 180;
    const int r  = (i % 180) / 30;
    const int cc = i % 30;
    const int h  = rowbase + r - 1;     // pad_l = pad_r = 1
    const int w  = colbase + cc - 1;
    if ((unsigned)h < HW && (unsigned)w < HW) {
      const float* g = Xb + ((size_t)c * HW + h) * HW + w;
      float* l = &Xlds[c * XC_STR + r * XR_STR + cc];
#if USE_ASYNC_LDS
      async_g2l_b32(g, l);
#else
      *l = *g;
#endif
    }
  }

  // ---------------- stage: weights, transposed to [d][oc'] ----------------
  const float* Wb = Wg + (size_t)ocbase * DTOT;
  for (int i = tid; i < OCB * DTOT; i += NTHREADS) {
    const int o = i / DTOT;
    const int d = i % DTOT;             // consecutive i -> consecutive d: coalesced reads
    const float* g = Wb + (size_t)o * DTOT + d;
    float* l = &Wlds[d * OCB + o];
#if USE_ASYNC_LDS
    async_g2l_b32(g, l);
#else
    *l = *g;
#endif
  }

#if USE_ASYNC_LDS
  wait_async0();
#endif
  __syncthreads();

  // ---------------- compute: 4 channels x 4 cols per thread ----------------
  float acc[4][4];
#pragma unroll
  for (int i = 0; i < 4; ++i)
#pragma unroll
    for (int j = 0; j < 4; ++j) acc[i][j] = 0.0f;

  const float* xbase = &Xlds[row * XR_STR + colb];
  const float* wbase = &Wlds[og * 4];

  for (int c = 0; c < CIN; ++c) {
    const float* xc = xbase + c * XC_STR;
#pragma unroll
    for (int kh = 0; kh < 3; ++kh) {
      const float* xr = xc + kh * XR_STR;
      const float4 xv0 = *reinterpret_cast<const float4*>(xr);      // ds_load_b128
      const float2 xv1 = *reinterpret_cast<const float2*>(xr + 4);  // ds_load_b64
      const float xs[6] = {xv0.x, xv0.y, xv0.z, xv0.w, xv1.x, xv1.y};

      const int d0 = c * 9 + kh * 3;
      const float4 w0 = *reinterpret_cast<const float4*>(wbase + (size_t)(d0 + 0) * OCB);
      const float4 w1 = *reinterpret_cast<const float4*>(wbase + (size_t)(d0 + 1) * OCB);
      const float4 w2 = *reinterpret_cast<const float4*>(wbase + (size_t)(d0 + 2) * OCB);
      const float wk0[4] = {w0.x, w0.y, w0.z, w0.w};
      const float wk1[4] = {w1.x, w1.y, w1.z, w1.w};
      const float wk2[4] = {w2.x, w2.y, w2.z, w2.w};

#pragma unroll
      for (int j = 0; j < 4; ++j) {
#pragma unroll
        for (int i = 0; i < 4; ++i) {
          const float t0 = xs[j + 0] - wk0[i];  // kw = 0
          const float t1 = xs[j + 1] - wk1[i];  // kw = 1
          const float t2 = xs[j + 2] - wk2[i];  // kw = 2
          // abs folds into VOP3 |src| modifiers; chain fuses into v_max3_num_f32
          acc[i][j] = fmaxf(fmaxf(fmaxf(acc[i][j], fabsf(t0)), fabsf(t1)), fabsf(t2));
        }
      }
    }
  }

  // ---------------- epilogue: + bias, float4 stores ----------------
  const int hout = rowbase + row;
  const int wout = colbase + colb;
#pragma unroll
  for (int i = 0; i < 4; ++i) {
    const int oc = ocbase + og * 4 + i;
    const float bv = Bias[oc];
    float4 r;
    r.x = acc[i][0] + bv;
    r.y = acc[i][1] + bv;
    r.z = acc[i][2] + bv;
    r.w = acc[i][3] + bv;
    *reinterpret_cast<float4*>(
        Out + (((size_t)b * COUT + oc) * HW + hout) * HW + wout) = r;
  }
}

extern "C" void kernel_launch(void* const* d_in, const int* in_sizes, int n_in,
                              void* d_out, int out_size, void* d_ws, size_t ws_size,
                              hipStream_t stream) {
  (void)in_sizes; (void)n_in; (void)d_ws; (void)ws_size; (void)out_size;
  const float* x    = (const float*)d_in[0];
  const float* w    = (const float*)d_in[1];
  const float* bias = (const float*)d_in[2];
  float* out        = (float*)d_out;

  // 120 KB dynamic LDS (> default 64 KB cap); gfx1250 WGP has 320 KB.
  (void)hipFuncSetAttribute(reinterpret_cast<const void*>(cheb_conv_kernel),
                            hipFuncAttributeMaxDynamicSharedMemorySize, LDS_BYTES);

  dim3 grid(8 /* 4 oc-blocks * 2 col-tiles */, 14 /* row tiles */, BATCH);
  dim3 block(NTHREADS);
  cheb_conv_kernel<<<grid, block, LDS_BYTES, stream>>>(x, w, bias, out);
}